// SPUTransformer_55267639165079
// MI455X (gfx1250) — compile-verified
//
#include <hip/hip_runtime.h>

// SPU bound-propagation: the reference's matmuls are all on diagonal matrices,
// so the entire _forward collapses to an independent per-row elementwise chain.
// N = 2048 rows, [lo, hi] per row, float32 in/out. Launch-latency bound
// (32 KB total traffic ~ 1.4 ns at 23.3 TB/s). One row per lane, wave32 blocks.

__device__ __forceinline__ float fast_rcp(float x) {
    return __builtin_amdgcn_rcpf(x);     // v_rcp_f32, ~1 ULP (clang builtin)
}

struct SpuTR {
    float sl, su;   // slopes  (lower, upper)
    float tl, tu;   // shifts  (lower, upper)
    float bl, bu;   // out bounds (lower, upper)
};

__device__ __forceinline__ float spu_f(float x) {
    // x >= 0 : x*x - 0.5
    // x <  0 : sigmoid(-x) - 1 = 1/(1+exp(x)) - 1 = -exp(x) / (1 + exp(x))
    float pos = fmaf(x, x, -0.5f);
    float e   = __expf(x);               // v_exp_f32 transcendental
    float neg = -e * fast_rcp(1.0f + e); // exp(x) in (0,1) for x<0: stable
    return (x >= 0.0f) ? pos : neg;
}

__device__ __forceinline__ SpuTR spu_transform(float l, float u) {
    float vl = spu_f(l);
    float vu = spu_f(u);
    float diff = u - l;

    bool neg   = (u <= 0.0f);
    bool pos   = (l >= 0.0f);
    bool cross = !(neg || pos);

    float s = (vu - vl) * fast_rcp(diff);

    float slope_u = (pos || cross) ? s : 0.0f;
    float slope_l = neg            ? s : 0.0f;

    float shift_u = (pos || cross) ? fmaf(-slope_u, u, vu) : 0.0f;
    float shift_l = neg            ? fmaf(-slope_l, u, vu) : 0.0f;

    bool swap = (s < 0.0f);
    float bl = swap ? vu : vl;
    float bu = swap ? vl : vu;
    bl = cross ? -0.5f : bl;

    shift_u = neg            ? bu : shift_u;
    shift_l = (pos || cross) ? bl : shift_l;

    SpuTR r;
    r.sl = slope_l; r.su = slope_u;
    r.tl = shift_l; r.tu = shift_u;
    r.bl = bl;      r.bu = bu;
    return r;
}

__global__ void __launch_bounds__(256)
spu_forward_kernel(const float* __restrict__ bounds,
                   float* __restrict__ out, int n_rows) {
    int i = blockIdx.x * blockDim.x + threadIdx.x;
    if (i >= n_rows) return;

    // gfx1250 prefetch path (global_prefetch_b8); harmless hint, no counter.
    __builtin_prefetch(bounds + 2 * i, 0, 0);

    float2 b = reinterpret_cast<const float2*>(bounds)[i];  // global_load_b64
    float lo = b.x;
    float hi = b.y;

    // First transform on the input bounds, second on its output bounds.
    SpuTR p = spu_transform(lo, hi);
    SpuTR t = spu_transform(p.bl, p.bu);

    // Diagonal algebra: UM @ diag(s_p_u) etc. -> elementwise products.
    float UBMd = t.su * p.su;
    float LBMd = t.sl * p.sl;
    float UBV  = fmaf(t.su, p.tu, t.tu);
    float LBV  = fmaf(t.sl, p.tl, t.tl);

    // clip(M,0,inf)@x + clip(M,-inf,0)@y + V, per diagonal element.
    float lower = fmaf(fmaxf(LBMd, 0.0f), lo, fmaf(fminf(LBMd, 0.0f), hi, LBV));
    float upper = fmaf(fmaxf(UBMd, 0.0f), hi, fmaf(fminf(UBMd, 0.0f), lo, UBV));

    // where(lower > b_t_l, lower, b_t_l) / where(upper < b_t_u, upper, b_t_u)
    float new_l = (lower > t.bl) ? lower : t.bl;
    float new_u = (upper < t.bu) ? upper : t.bu;

    reinterpret_cast<float2*>(out)[i] = make_float2(new_l, new_u);  // global_store_b64
}

extern "C" void kernel_launch(void* const* d_in, const int* in_sizes, int n_in,
                              void* d_out, int out_size, void* d_ws, size_t ws_size,
                              hipStream_t stream) {
    const float* bounds = reinterpret_cast<const float*>(d_in[0]);
    float* out = reinterpret_cast<float*>(d_out);

    int n_rows = in_sizes[0] / 2;           // [N, 2] flattened
    const int block = 256;                  // 8 wave32 waves per workgroup
    int grid = (n_rows + block - 1) / block;

    spu_forward_kernel<<<grid, block, 0, stream>>>(bounds, out, n_rows);
}